// GNN_16432544875301
// MI455X (gfx1250) — compile-verified
//
#include <hip/hip_runtime.h>
#include <hip/hip_bf16.h>

typedef __attribute__((ext_vector_type(2))) float v2f;
typedef __attribute__((ext_vector_type(8))) float v8f;

#define NEG_SLOPE 0.2f

// ---------------- helpers ----------------

// Float atomic max via ordered-int trick (native global_atomic_max/min):
// non-negative floats compare correctly as signed ints; negative floats
// compare inversely as unsigned ints. Init value must be -inf (0xFF800000).
__device__ __forceinline__ void atomic_max_f32(float* addr, float v) {
  if (v >= 0.0f) atomicMax((int*)addr, __float_as_int(v));
  else           atomicMin((unsigned int*)addr, __float_as_uint(v));
}

// edge_index is [2,E] flat; e in [0, E+N): last N are self loops.
__device__ __forceinline__ void edge_nodes(const int* __restrict__ ei, int E,
                                           int e, int& s, int& d) {
  if (e < E) { s = ei[e]; d = ei[E + e]; }
  else       { s = e - E; d = s; }
}

__global__ void fill_f32(float* p, float v, long long n) {
  long long i = (long long)blockIdx.x * blockDim.x + threadIdx.x;
  if (i < n) p[i] = v;
}

// ---------------- layer-1 GEMM: h1[N,128] = x[N,128] @ W1[128,128] ----------------
// 1 wave -> 16 rows x 128 cols via 8x V_WMMA_F32_16X16X4_F32, K-loop of 32.
// W staged in LDS K-pair interleaved: lW2[(k/2)*128 + col] = (W[k][col], W[k+1][col])
// so each B operand is a single aligned ds_load_b64 into a VGPR pair.
__global__ void __launch_bounds__(256) gemm1_wmma(
    const float* __restrict__ x, const float* __restrict__ W,
    float* __restrict__ h1, int n) {
  __shared__ v2f lW2[64 * 128];  // 64 KB of 320 KB WGP LDS
  const int t = threadIdx.x;
  for (int i = t; i < 64 * 128; i += 256) {
    int kp = i >> 7, col = i & 127;
    v2f w;
    w.x = W[(2 * kp) * 128 + col];
    w.y = W[(2 * kp + 1) * 128 + col];
    lW2[i] = w;
  }
  __syncthreads();

  const int lane = t & 31, wave = t >> 5;
  const int row0 = blockIdx.x * 128 + wave * 16;
  if (row0 >= n) return;  // uniform per wave
  const int half = lane >> 4;   // 0: K+0/1 (A) / M+0..7 (D); 1: K+2/3 / M+8..15
  const int m = lane & 15;
  int arow = row0 + m; if (arow > n - 1) arow = n - 1;  // branchless row clamp
  const float* xr = x + (size_t)arow * 128;

  const v8f vzero = {0, 0, 0, 0, 0, 0, 0, 0};
  v8f acc[8];
#pragma unroll
  for (int cb = 0; cb < 8; ++cb) acc[cb] = vzero;

  for (int k0 = 0; k0 < 128; k0 += 4) {
    // A 16x4: lane<16 holds (K0,K1) of row m; lane>=16 holds (K2,K3)
    v2f a = *(const v2f*)(xr + k0 + half * 2);
    const v2f* wp = lW2 + (k0 / 2 + half) * 128 + m;
#pragma unroll
    for (int cb = 0; cb < 8; ++cb) {
      // B 4x16: pair (row K0+half*2, row K0+half*2+1) at column cb*16+m
      v2f b = wp[cb * 16];
      acc[cb] = __builtin_amdgcn_wmma_f32_16x16x4_f32(
          false, a, false, b, (short)0, acc[cb], false, false);
    }
  }

  // D 16x16: VGPR r -> M = r + half*8, N = m
#pragma unroll
  for (int cb = 0; cb < 8; ++cb)
#pragma unroll
    for (int r = 0; r < 8; ++r) {
      int row = row0 + half * 8 + r;
      if (row < n) h1[(size_t)row * 128 + cb * 16 + m] = acc[cb][r];
    }
}

// ---------------- layer-2 GEMM: h2[N,16] = relu(agg1+b1) @ W2[128,16] ----------------
__global__ void __launch_bounds__(256) gemm2_wmma(
    const float* __restrict__ agg1, const float* __restrict__ b1,
    const float* __restrict__ W2, float* __restrict__ h2, int n) {
  __shared__ v2f lW2[64 * 16];  // 8 KB, pair-interleaved
  const int t = threadIdx.x;
  for (int i = t; i < 64 * 16; i += 256) {
    int kp = i >> 4, col = i & 15;
    v2f w;
    w.x = W2[(2 * kp) * 16 + col];
    w.y = W2[(2 * kp + 1) * 16 + col];
    lW2[i] = w;
  }
  __syncthreads();

  const int lane = t & 31, wave = t >> 5;
  const int row0 = blockIdx.x * 128 + wave * 16;
  if (row0 >= n) return;
  const int half = lane >> 4, m = lane & 15;
  int arow = row0 + m; if (arow > n - 1) arow = n - 1;
  const float* ar = agg1 + (size_t)arow * 128;

  v8f acc = {0, 0, 0, 0, 0, 0, 0, 0};
  for (int k0 = 0; k0 < 128; k0 += 4) {
    int ka = k0 + half * 2;
    float a0 = ar[ka] + b1[ka];         a0 = a0 > 0.f ? a0 : 0.f;  // fused relu(agg1+b1)
    float a1 = ar[ka + 1] + b1[ka + 1]; a1 = a1 > 0.f ? a1 : 0.f;
    v2f a; a.x = a0; a.y = a1;
    v2f b = lW2[(k0 / 2 + half) * 16 + m];
    acc = __builtin_amdgcn_wmma_f32_16x16x4_f32(
        false, a, false, b, (short)0, acc, false, false);
  }
#pragma unroll
  for (int r = 0; r < 8; ++r) {
    int row = row0 + half * 8 + r;
    if (row < n) h2[(size_t)row * 16 + m] = acc[r];
  }
}

// ---------------- attention logits ----------------
__global__ void attn1(const float* __restrict__ h1, const float* __restrict__ as,
                      const float* __restrict__ ad, float* __restrict__ als,
                      float* __restrict__ ald, int n) {
  int i = blockIdx.x * blockDim.x + threadIdx.x;
  if (i >= n) return;
  const float* hr = h1 + (size_t)i * 128;
  float s0 = 0, s1 = 0, d0 = 0, d1 = 0;
  for (int c = 0; c < 64; ++c) {
    float v0 = hr[c], v1 = hr[64 + c];
    s0 += v0 * as[c]; s1 += v1 * as[64 + c];
    d0 += v0 * ad[c]; d1 += v1 * ad[64 + c];
  }
  als[i * 2 + 0] = s0; als[i * 2 + 1] = s1;
  ald[i * 2 + 0] = d0; ald[i * 2 + 1] = d1;
}

__global__ void attn2(const float* __restrict__ h2, const float* __restrict__ as,
                      const float* __restrict__ ad, float* __restrict__ als,
                      float* __restrict__ ald, int n) {
  int i = blockIdx.x * blockDim.x + threadIdx.x;
  if (i >= n) return;
  const float* hr = h2 + (size_t)i * 16;
  float s = 0, d = 0;
  for (int c = 0; c < 16; ++c) { s += hr[c] * as[c]; d += hr[c] * ad[c]; }
  als[i] = s; ald[i] = d;
}

// ---------------- edge passes, layer 1 (2 heads) ----------------
__global__ void edge_max1(const int* __restrict__ ei, int E, int EA,
                          const float* __restrict__ als, const float* __restrict__ ald,
                          float* __restrict__ emax) {
  int e = blockIdx.x * blockDim.x + threadIdx.x;
  if (e >= EA) return;
  int s, d; edge_nodes(ei, E, e, s, d);
#pragma unroll
  for (int h = 0; h < 2; ++h) {
    float v = als[s * 2 + h] + ald[d * 2 + h];
    v = v > 0.f ? v : NEG_SLOPE * v;
    atomic_max_f32(&emax[d * 2 + h], v);
  }
}

__global__ void edge_sum1(const int* __restrict__ ei, int E, int EA,
                          const float* __restrict__ als, const float* __restrict__ ald,
                          const float* __restrict__ emax, float* __restrict__ denom) {
  int e = blockIdx.x * blockDim.x + threadIdx.x;
  if (e >= EA) return;
  int s, d; edge_nodes(ei, E, e, s, d);
#pragma unroll
  for (int h = 0; h < 2; ++h) {
    float v = als[s * 2 + h] + ald[d * 2 + h];
    v = v > 0.f ? v : NEG_SLOPE * v;
    atomicAdd(&denom[d * 2 + h], __expf(v - emax[d * 2 + h]));
  }
}

// wave per edge; lane covers 4 channels (lanes 0-15 head0, 16-31 head1)
__global__ void __launch_bounds__(256) edge_agg1(
    const int* __restrict__ ei, int E, int EA,
    const float* __restrict__ als, const float* __restrict__ ald,
    const float* __restrict__ emax, const float* __restrict__ denom,
    const float* __restrict__ h1, float* __restrict__ agg) {
  int gid = blockIdx.x * blockDim.x + threadIdx.x;
  int e = gid >> 5;
  if (e >= EA) return;
  int lane = gid & 31;
  int s, d; edge_nodes(ei, E, e, s, d);
  int h = lane >> 4;
  float v = als[s * 2 + h] + ald[d * 2 + h];
  v = v > 0.f ? v : NEG_SLOPE * v;
  float alpha = __expf(v - emax[d * 2 + h]) / (denom[d * 2 + h] + 1e-16f);
  int c0 = h * 64 + (lane & 15) * 4;
  float4 hv = *(const float4*)(h1 + (size_t)s * 128 + c0);
  float* ap = agg + (size_t)d * 128 + c0;
  atomicAdd(ap + 0, hv.x * alpha);
  atomicAdd(ap + 1, hv.y * alpha);
  atomicAdd(ap + 2, hv.z * alpha);
  atomicAdd(ap + 3, hv.w * alpha);
}

// ---------------- edge passes, layer 2 (1 head, 16 ch) ----------------
__global__ void edge_max2(const int* __restrict__ ei, int E, int EA,
                          const float* __restrict__ als, const float* __restrict__ ald,
                          float* __restrict__ emax) {
  int e = blockIdx.x * blockDim.x + threadIdx.x;
  if (e >= EA) return;
  int s, d; edge_nodes(ei, E, e, s, d);
  float v = als[s] + ald[d];
  v = v > 0.f ? v : NEG_SLOPE * v;
  atomic_max_f32(&emax[d], v);
}

__global__ void edge_sum2(const int* __restrict__ ei, int E, int EA,
                          const float* __restrict__ als, const float* __restrict__ ald,
                          const float* __restrict__ emax, float* __restrict__ denom) {
  int e = blockIdx.x * blockDim.x + threadIdx.x;
  if (e >= EA) return;
  int s, d; edge_nodes(ei, E, e, s, d);
  float v = als[s] + ald[d];
  v = v > 0.f ? v : NEG_SLOPE * v;
  atomicAdd(&denom[d], __expf(v - emax[d]));
}

// 16 threads per edge, one channel each
__global__ void __launch_bounds__(256) edge_agg2(
    const int* __restrict__ ei, int E, int EA,
    const float* __restrict__ als, const float* __restrict__ ald,
    const float* __restrict__ emax, const float* __restrict__ denom,
    const float* __restrict__ h2, float* __restrict__ agg) {
  int gid = blockIdx.x * blockDim.x + threadIdx.x;
  int e = gid >> 4;
  if (e >= EA) return;
  int c = gid & 15;
  int s, d; edge_nodes(ei, E, e, s, d);
  float v = als[s] + ald[d];
  v = v > 0.f ? v : NEG_SLOPE * v;
  float alpha = __expf(v - emax[d]) / (denom[d] + 1e-16f);
  atomicAdd(&agg[(size_t)d * 16 + c], h2[(size_t)s * 16 + c] * alpha);
}

__global__ void finalize(const float* __restrict__ agg2, const float* __restrict__ b2,
                         float* __restrict__ out, int n16) {
  int i = blockIdx.x * blockDim.x + threadIdx.x;
  if (i >= n16) return;
  out[i] = agg2[i] + b2[i & 15];
}

// ---------------- launch ----------------
extern "C" void kernel_launch(void* const* d_in, const int* in_sizes, int n_in,
                              void* d_out, int out_size, void* d_ws, size_t ws_size,
                              hipStream_t stream) {
  const float* x   = (const float*)d_in[0];
  const int*   ei  = (const int*)d_in[1];
  const float* W1  = (const float*)d_in[2];
  const float* a1s = (const float*)d_in[3];
  const float* a1d = (const float*)d_in[4];
  const float* b1  = (const float*)d_in[5];
  const float* W2  = (const float*)d_in[6];
  const float* a2s = (const float*)d_in[7];
  const float* a2d = (const float*)d_in[8];
  const float* b2  = (const float*)d_in[9];
  float* out = (float*)d_out;

  const int n  = in_sizes[0] / 128;
  const int E  = in_sizes[1] / 2;
  const int EA = E + n;  // + self loops

  float* ws = (float*)d_ws;
  size_t off = 0;
  float* h1   = ws + off; off += (size_t)n * 128;
  float* agg1 = ws + off; off += (size_t)n * 128;
  float* h2   = ws + off; off += (size_t)n * 16;
  float* agg2 = ws + off; off += (size_t)n * 16;
  float* al1s = ws + off; off += (size_t)n * 2;
  float* al1d = ws + off; off += (size_t)n * 2;
  float* em1  = ws + off; off += (size_t)n * 2;
  float* dn1  = ws + off; off += (size_t)n * 2;
  float* al2s = ws + off; off += (size_t)n;
  float* al2d = ws + off; off += (size_t)n;
  float* em2  = ws + off; off += (size_t)n;
  float* dn2  = ws + off; off += (size_t)n;

  hipMemsetAsync(agg1, 0, (size_t)n * 128 * sizeof(float), stream);
  hipMemsetAsync(agg2, 0, (size_t)n * 16 * sizeof(float), stream);
  hipMemsetAsync(dn1,  0, (size_t)n * 2 * sizeof(float), stream);
  hipMemsetAsync(dn2,  0, (size_t)n * sizeof(float), stream);
  fill_f32<<<(n * 2 + 255) / 256, 256, 0, stream>>>(em1, -__builtin_inff(), (long long)n * 2);
  fill_f32<<<(n + 255) / 256, 256, 0, stream>>>(em2, -__builtin_inff(), (long long)n);

  // layer 1
  gemm1_wmma<<<(n + 127) / 128, 256, 0, stream>>>(x, W1, h1, n);
  attn1<<<(n + 255) / 256, 256, 0, stream>>>(h1, a1s, a1d, al1s, al1d, n);
  edge_max1<<<(EA + 255) / 256, 256, 0, stream>>>(ei, E, EA, al1s, al1d, em1);
  edge_sum1<<<(EA + 255) / 256, 256, 0, stream>>>(ei, E, EA, al1s, al1d, em1, dn1);
  {
    long long tt = (long long)EA * 32;
    edge_agg1<<<(unsigned)((tt + 255) / 256), 256, 0, stream>>>(
        ei, E, EA, al1s, al1d, em1, dn1, h1, agg1);
  }

  // layer 2 (relu(agg1+b1) fused into GEMM A-loads)
  gemm2_wmma<<<(n + 127) / 128, 256, 0, stream>>>(agg1, b1, W2, h2, n);
  attn2<<<(n + 255) / 256, 256, 0, stream>>>(h2, a2s, a2d, al2s, al2d, n);
  edge_max2<<<(EA + 255) / 256, 256, 0, stream>>>(ei, E, EA, al2s, al2d, em2);
  edge_sum2<<<(EA + 255) / 256, 256, 0, stream>>>(ei, E, EA, al2s, al2d, em2, dn2);
  {
    long long tt = (long long)EA * 16;
    edge_agg2<<<(unsigned)((tt + 255) / 256), 256, 0, stream>>>(
        ei, E, EA, al2s, al2d, em2, dn2, h2, agg2);
  }

  finalize<<<(n * 16 + 255) / 256, 256, 0, stream>>>(agg2, b2, out, n * 16);
}